// ManualMaxPool2d_69861938037161
// MI455X (gfx1250) — compile-verified
//
#include <hip/hip_runtime.h>
#include <hip/hip_bf16.h>

// Max-pool 2x2, stride 2, VALID padding, NCHW fp32.
// In:  (32, 96, 224, 224)  -> 154,140,672 floats (616.6 MB)
// Out: (32, 96, 112, 112)  ->  38,535,168 floats (154.1 MB)
//
// Pure HBM-bandwidth kernel: 771 MB total traffic -> ~33 us floor @ 23.3 TB/s.
// Strategy (round 2 — remove all integer-division index math):
//  - geometry-matched launch: block=(28,8) [224 thr = 7 wave32],
//    grid=(14, 3072). Exact cover of (28 quads/row, 112 rows, 3072 planes);
//    no %, no /, no bounds checks.
//  - one thread -> 4 consecutive output pixels: 2x b128 NT loads from input
//    row 2y, 2x from row 2y+1, 1x b128 NT store (64 B in / 16 B out).
//  - all accesses 16-B aligned; lanes 0..27 of a wave cover one contiguous
//    896-B input row segment (fully coalesced under wave32).
//  - TH=NT cache hints (gfx1250 temporal-hint path): 616 MB input stream
//    exceeds the 192 MB L2 and is never re-read.

typedef float v4f __attribute__((ext_vector_type(4)));

namespace {
constexpr int H_IN  = 224;
constexpr int W_IN  = 224;
constexpr int H_OUT = 112;
constexpr int W_OUT = 112;
constexpr int N_IMG = 32 * 96;            // 3072 flattened N*C planes
constexpr int QUADS_PER_ROW = W_OUT / 4;  // 28 -> blockDim.x
constexpr int ROWS_PER_BLOCK = 8;         // -> blockDim.y (28*8 = 224 threads)
constexpr int ROW_GROUPS = H_OUT / ROWS_PER_BLOCK;  // 14 -> gridDim.x
}

__global__ void maxpool2x2_nchw_kernel(const float* __restrict__ in,
                                       float* __restrict__ out) {
    const int q   = threadIdx.x;                          // 0..27  quad in row
    const int oy  = blockIdx.x * ROWS_PER_BLOCK + threadIdx.y; // 0..111 out row
    const int img = blockIdx.y;                           // 0..3071 NC plane

    // Input: rows 2*oy and 2*oy+1, x range [8*q, 8*q+8)
    const float* base = in + (size_t)img * (H_IN * W_IN)
                           + (size_t)(2 * oy) * W_IN
                           + (size_t)q * 8;

    const v4f* r0 = reinterpret_cast<const v4f*>(base);          // row 2y
    const v4f* r1 = reinterpret_cast<const v4f*>(base + W_IN);   // row 2y+1

    // 4x 128-bit non-temporal loads (streamed exactly once)
    v4f a0 = __builtin_nontemporal_load(r0);
    v4f a1 = __builtin_nontemporal_load(r0 + 1);
    v4f b0 = __builtin_nontemporal_load(r1);
    v4f b1 = __builtin_nontemporal_load(r1 + 1);

    // 2x2 disjoint-window maxima
    v4f o;
    o[0] = fmaxf(fmaxf(a0[0], a0[1]), fmaxf(b0[0], b0[1]));
    o[1] = fmaxf(fmaxf(a0[2], a0[3]), fmaxf(b0[2], b0[3]));
    o[2] = fmaxf(fmaxf(a1[0], a1[1]), fmaxf(b1[0], b1[1]));
    o[3] = fmaxf(fmaxf(a1[2], a1[3]), fmaxf(b1[2], b1[3]));

    float* op = out + (size_t)img * (H_OUT * W_OUT)
                    + (size_t)oy * W_OUT
                    + (size_t)q * 4;

    // 128-bit non-temporal store
    __builtin_nontemporal_store(o, reinterpret_cast<v4f*>(op));
}

extern "C" void kernel_launch(void* const* d_in, const int* in_sizes, int n_in,
                              void* d_out, int out_size, void* d_ws, size_t ws_size,
                              hipStream_t stream) {
    (void)in_sizes; (void)n_in; (void)out_size; (void)d_ws; (void)ws_size;

    const float* x = reinterpret_cast<const float*>(d_in[0]);
    float* out     = reinterpret_cast<float*>(d_out);

    dim3 block(QUADS_PER_ROW, ROWS_PER_BLOCK, 1);  // (28, 8) = 224 = 7 waves
    dim3 grid(ROW_GROUPS, N_IMG, 1);               // (14, 3072)

    maxpool2x2_nchw_kernel<<<grid, block, 0, stream>>>(x, out);
}